// Baseline_4595615006757
// MI455X (gfx1250) — compile-verified
//
#include <hip/hip_runtime.h>
#include <hip/hip_bf16.h>

// Problem constants (match reference)
#define NROWS 8192
#define DDIM  1024
#define EEDGE 8192
#define KNBR  16

typedef float v2f __attribute__((ext_vector_type(2)));
typedef float v4f __attribute__((ext_vector_type(4)));
typedef float v8f __attribute__((ext_vector_type(8)));

// ---------------------------------------------------------------------------
// Kernel 1: s = sent_feat @ [w1 w2]  via V_WMMA_F32_16X16X4_F32 (exact fp32).
// One wave (32 lanes) per 16-row tile. 1024/4 = 256 K-steps, 2 accumulators.
//
// A (16x4 f32) lane layout (ISA 7.12.2): lanes 0-15: M=lane, VGPR0=K0,VGPR1=K1
//                                        lanes 16-31: M=lane-16, K2,K3
// B (4x16 f32): lanes 0-15: N=lane, VGPR0=K0,VGPR1=K1; lanes 16-31: K2,K3
// D (16x16 f32): VGPR r: lanes 0-15 -> M=r,N=lane ; lanes 16-31 -> M=r+8
// ---------------------------------------------------------------------------
__global__ __launch_bounds__(32) void matvec_wmma_kernel(
    const float* __restrict__ feat, const float* __restrict__ W,
    float* __restrict__ s1, float* __restrict__ s2) {
  const int lane = threadIdx.x;          // 0..31 (wave32)
  const int m0   = blockIdx.x * 16;
  const int row  = lane & 15;            // A-matrix row for this lane
  const int kofs = (lane >> 4) * 2;      // 0 or 2: which K-pair this lane holds
  const int ncol = lane & 15;            // B-matrix column for this lane

  const float* __restrict__ arow = feat + (size_t)(m0 + row) * DDIM + kofs;
  // B column 0 = w1 chunk, column 1 = w2 chunk, columns 2..15 = 0
  const float* __restrict__ wsel = (ncol == 0) ? (W + kofs) : (W + DDIM + kofs);
  const bool bactive = (ncol < 2);

  v8f acc0 = {};
  v8f acc1 = {};
  for (int k = 0; k < DDIM; k += 8) {
    v2f a0, a1;
    a0[0] = arow[k];
    a0[1] = arow[k + 1];
    a1[0] = arow[k + 4];
    a1[1] = arow[k + 5];
    v2f b0 = {0.0f, 0.0f};
    v2f b1 = {0.0f, 0.0f};
    if (bactive) {
      b0[0] = wsel[k];
      b0[1] = wsel[k + 1];
      b1[0] = wsel[k + 4];
      b1[1] = wsel[k + 5];
    }
    acc0 = __builtin_amdgcn_wmma_f32_16x16x4_f32(false, a0, false, b0,
                                                 (short)0, acc0, false, false);
    acc1 = __builtin_amdgcn_wmma_f32_16x16x4_f32(false, a1, false, b1,
                                                 (short)0, acc1, false, false);
  }
  v8f acc = acc0 + acc1;   // element-wise add == matrix add (identical layouts)

  // Column 0 (s1) lives in lanes 0 (M=0..7) and 16 (M=8..15);
  // column 1 (s2) in lanes 1 and 17.
  const int mbase = m0 + ((lane >> 4) ? 8 : 0);
  if (ncol == 0) {
#pragma unroll
    for (int r = 0; r < 8; ++r) s1[mbase + r] = acc[r];
  } else if (ncol == 1) {
#pragma unroll
    for (int r = 0; r < 8; ++r) s2[mbase + r] = acc[r];
  }
}

// ---------------------------------------------------------------------------
// Kernel 2: all_score[i,j] = s1[i] + s2[j] + b   (256 MB of stores — the
// bandwidth bottleneck). v4f non-temporal stores; each block reuses its s2
// slice across 8 rows.  grid = (2, 1024), block = 256 threads.
// Per thread: 4 column-chunks x 8 rows = 32 x 16B stores.
// ---------------------------------------------------------------------------
__global__ __launch_bounds__(256) void all_score_kernel(
    const float* __restrict__ s1, const float* __restrict__ s2,
    const float* __restrict__ bptr, float* __restrict__ out) {
  const int i0 = blockIdx.y * 8;                       // first of 8 rows
  const int c0 = blockIdx.x * 1024 + threadIdx.x;      // column in v4f units
  const float bb = bptr[0];

  float a[8];
#pragma unroll
  for (int r = 0; r < 8; ++r) a[r] = s1[i0 + r] + bb;

  const v4f* __restrict__ s2v = (const v4f*)s2;
  v4f* __restrict__ ov = (v4f*)out;

#pragma unroll
  for (int u = 0; u < 4; ++u) {
    const int c = c0 + u * 256;
    const v4f v = s2v[c];
#pragma unroll
    for (int r = 0; r < 8; ++r) {
      v4f o = v;
      o[0] += a[r]; o[1] += a[r]; o[2] += a[r]; o[3] += a[r];
      __builtin_nontemporal_store(o, &ov[(size_t)(i0 + r) * (NROWS / 4) + c]);
    }
  }
}

// ---------------------------------------------------------------------------
// Kernel 3: pair_score[e,j] = s1[edge[e,0]] + s2[edge[e,1+j]] + b
// s1/s2 are 32 KB each -> L2-resident gathers. One thread per output.
// ---------------------------------------------------------------------------
__global__ __launch_bounds__(256) void pair_score_kernel(
    const float* __restrict__ s1, const float* __restrict__ s2,
    const float* __restrict__ bptr, const int* __restrict__ edge,
    float* __restrict__ out) {
  const int idx = blockIdx.x * blockDim.x + threadIdx.x;
  if (idx >= EEDGE * (KNBR - 1)) return;
  const int e = idx / (KNBR - 1);
  const int j = idx - e * (KNBR - 1);
  const int ci = edge[e * KNBR];
  const int ni = edge[e * KNBR + 1 + j];
  out[idx] = s1[ci] + s2[ni] + bptr[0];
}

extern "C" void kernel_launch(void* const* d_in, const int* in_sizes, int n_in,
                              void* d_out, int out_size, void* d_ws, size_t ws_size,
                              hipStream_t stream) {
  (void)in_sizes; (void)n_in; (void)out_size; (void)ws_size;
  const float* feat = (const float*)d_in[0];   // (N, D) fp32
  const float* W    = (const float*)d_in[1];   // (2D,)  fp32
  const float* b    = (const float*)d_in[2];   // (1,)   fp32
  const int*   edge = (const int*)d_in[3];     // (E, K) int32

  float* pair = (float*)d_out;                       // E*(K-1) = 122880 floats
  float* all  = pair + (size_t)EEDGE * (KNBR - 1);   // N*N floats

  float* s1 = (float*)d_ws;                          // N floats
  float* s2 = s1 + NROWS;                            // N floats

  // 1) s1, s2 via fp32 WMMA: 512 waves of 16 rows each.
  matvec_wmma_kernel<<<NROWS / 16, 32, 0, stream>>>(feat, W, s1, s2);

  // 2) 256 MB broadcast-add store (the bandwidth bottleneck).
  all_score_kernel<<<dim3(2, NROWS / 8), 256, 0, stream>>>(s1, s2, b, all);

  // 3) tiny gather for pair_score.
  const int npair = EEDGE * (KNBR - 1);
  pair_score_kernel<<<(npair + 255) / 256, 256, 0, stream>>>(s1, s2, b, edge, pair);
}